// HORNSeqLayer_52725018525860
// MI455X (gfx1250) — compile-verified
//
#include <hip/hip_runtime.h>
#include <math.h>

typedef __attribute__((ext_vector_type(2))) float v2f;
typedef __attribute__((ext_vector_type(8))) float v8f;
typedef __attribute__((ext_vector_type(4))) unsigned int v4u;
typedef __attribute__((ext_vector_type(8))) int v8i;
typedef __attribute__((ext_vector_type(4))) int v4i;

#define N_INPUT  256
#define N_HIDDEN 512
#define T_STEPS  1024
#define BATCH    64
#define DT_F     0.1f
#define GAIN_REC 0.044194173824159216f   // 1/sqrt(512)
#define NWG_REC  16

// -------------------------------------------------------------------------
// Phase 0: reset the global barrier state (d_ws is poisoned, never restored)
// -------------------------------------------------------------------------
__global__ void horn_init(int* __restrict__ bar) {
  bar[0] = 0;   // arrival count
  bar[1] = 0;   // phase
}

// -------------------------------------------------------------------------
// Phase 1: ext[T*B, H] = inputs[T*B, 256] @ W_ih[256, 512] + b_ih
// One 16x16 output tile per wave via V_WMMA_F32_16X16X4_F32.
// Block = 256 threads = 8 waves = 64x32 output block. Memory-bound:
// 64MB in + 134MB out @ 23.3 TB/s ~ 8.5us; W_ih (512KB) stays L2-resident.
// -------------------------------------------------------------------------
__global__ void horn_ext_gemm(const float* __restrict__ inp,
                              const float* __restrict__ Wih,
                              const float* __restrict__ bih,
                              float* __restrict__ ext) {
  const int bcol  = blockIdx.x & 15;        // 16 column blocks of 32
  const int brow  = blockIdx.x >> 4;        // 1024 row blocks of 64
  const int lane  = threadIdx.x & 31;
  const int wv    = threadIdx.x >> 5;       // 0..7
  const int mTile = wv >> 1;                // 0..3
  const int nTile = wv & 1;                 // 0..1
  const int half  = lane >> 4;              // 0 | 1
  const int lm    = lane & 15;

  const int row0 = brow * 64 + mTile * 16;
  const int n    = bcol * 32 + nTile * 16 + lm;   // this lane's output column

  // A fragment source: row (row0+lm), k offset 2*half (layout of 16x4 f32 A)
  const float* Arow = inp + (size_t)(row0 + lm) * N_INPUT + 2 * half;
  // B fragment source: W_ih[k + 2*half][n]
  const float* Bcol = Wih + (size_t)(2 * half) * N_HIDDEN + n;

  v8f acc = {0.f, 0.f, 0.f, 0.f, 0.f, 0.f, 0.f, 0.f};
#pragma unroll 4
  for (int k0 = 0; k0 < N_INPUT; k0 += 4) {
    v2f a, b;
    a.x = Arow[k0 + 0];
    a.y = Arow[k0 + 1];
    b.x = Bcol[(size_t)(k0 + 0) * N_HIDDEN];
    b.y = Bcol[(size_t)(k0 + 1) * N_HIDDEN];
    acc = __builtin_amdgcn_wmma_f32_16x16x4_f32(false, a, false, b,
                                                (short)0, acc, false, false);
  }

  const float bias = bih[n];
#pragma unroll
  for (int r = 0; r < 8; ++r) {
    const int row = row0 + r + 8 * half;   // C layout: lane half selects M+8
    ext[(size_t)row * N_HIDDEN + n] = acc[r] + bias;
  }
}

// -------------------------------------------------------------------------
// Grid-wide phase barrier over NWG_REC persistent workgroups
// -------------------------------------------------------------------------
__device__ __forceinline__ void grid_barrier(int* __restrict__ bar, int nwg) {
  __syncthreads();
  if (threadIdx.x == 0) {
    const int ph = __hip_atomic_load(&bar[1], __ATOMIC_ACQUIRE,
                                     __HIP_MEMORY_SCOPE_AGENT);
    const int arrived = __hip_atomic_fetch_add(&bar[0], 1, __ATOMIC_ACQ_REL,
                                               __HIP_MEMORY_SCOPE_AGENT);
    if (arrived == nwg - 1) {
      __hip_atomic_store(&bar[0], 0, __ATOMIC_RELAXED, __HIP_MEMORY_SCOPE_AGENT);
      __hip_atomic_fetch_add(&bar[1], 1, __ATOMIC_RELEASE,
                             __HIP_MEMORY_SCOPE_AGENT);
    } else {
      while (__hip_atomic_load(&bar[1], __ATOMIC_ACQUIRE,
                               __HIP_MEMORY_SCOPE_AGENT) == ph) {
        __builtin_amdgcn_s_sleep(1);
      }
    }
  }
  __syncthreads();
}

// -------------------------------------------------------------------------
// Phase 2: persistent recurrence. 16 WGs x 256 threads; WG w owns hidden
// columns [w*32, w*32+32). The W_hh slice (512x32 f32 = 64KB) is DMA'd into
// LDS by the Tensor Data Mover; local x/y slices (8KB each) also in LDS.
// y ping-pongs through global d_ws between steps with a grid barrier.
// -------------------------------------------------------------------------
__global__ void horn_recur(float* __restrict__ out,            // [T,B,H] ext in / x out
                           const float* __restrict__ Whh,      // [512,512]
                           const float* __restrict__ bhh,
                           const float* __restrict__ alp,
                           const float* __restrict__ omg,
                           const float* __restrict__ gam,
                           const float* __restrict__ vco,
                           float* __restrict__ ybuf,           // [2][B*H] ping-pong
                           int* __restrict__ bar) {
  extern __shared__ float smem[];
  float* sW = smem;                    // 512*32 = 16384 floats (LDS offset 0)
  float* sX = smem + 512 * 32;         // 64*32
  float* sY = sX + 64 * 32;            // 64*32

  const int wg    = blockIdx.x;        // 0..15
  const int tid   = threadIdx.x;
  const int lane  = tid & 31;
  const int wv    = tid >> 5;          // 0..7 -> one 16x16 tile each
  const int mTile = wv >> 1;           // 0..3 (batch rows)
  const int nTile = wv & 1;            // 0..1 (16-col halves of the 32-col slice)
  const int half  = lane >> 4;
  const int lm    = lane & 15;
  const int m0    = mTile * 16;
  const int nloc  = nTile * 16 + lm;   // 0..31 within slice (B/C lane column)
  const int h     = wg * 32 + nloc;    // global hidden index

  // ---- TDM: DMA the 512x32 W_hh slice into LDS (wave 0 issues once) ----
  if (wv == 0) {
    const unsigned long long gaddr = (unsigned long long)(Whh + wg * 32);
    v4u g0;
    g0[0] = 1u;                                    // count=1 (valid descriptor)
    g0[1] = 0u;                                    // lds_addr = 0 (sW base)
    g0[2] = (unsigned int)(gaddr & 0xFFFFFFFFull); // global_addr[31:0]
    g0[3] = (unsigned int)((gaddr >> 32) & 0x1FFFFFFull) | (2u << 30); // addr[56:32] | type=2

    v8i g1;
    g1[0] = (2 << 16);        // workgroup_mask=0 | data_size=2 (4 bytes)
    g1[1] = (512 << 16);      // atomic_barrier_addr=0 | tensor_dim0[15:0]=512
    g1[2] = (512 << 16);      // tensor_dim0[31:16]=0 | tensor_dim1[15:0]=512
    g1[3] = (32 << 16);       // tensor_dim1[31:16]=0 | tile_dim0=32
    g1[4] = 512;              // tile_dim1=512 | tile_dim2=0
    g1[5] = 512;              // tensor_dim0_stride[31:0]=512
    g1[6] = 0;                // stride0[47:32]=0 | tensor_dim1_stride[15:0]=0
    g1[7] = 0;                // tensor_dim1_stride[47:16]=0

    v4i g2 = {0, 0, 0, 0};    // 2-D tile: groups 2/3 unused
    v4i g3 = {0, 0, 0, 0};
    v8i g4 = {0, 0, 0, 0, 0, 0, 0, 0};   // 6-arg toolchain variant: extra group
    __builtin_amdgcn_tensor_load_to_lds(g0, g1, g2, g3, g4, 0);
  }

  // Zero local state and publish y0 = 0 for our columns (overlaps the DMA)
  for (int idx = tid; idx < 64 * 32; idx += 256) {
    sX[idx] = 0.f;
    sY[idx] = 0.f;
    const int b = idx >> 5, j = idx & 31;
    ybuf[(size_t)b * N_HIDDEN + wg * 32 + j] = 0.f;
  }

  // Per-lane column-constant coefficients (one hidden column per lane)
  const float bh  = bhh[h];
  const float al  = alp[h];
  const float om  = omg[h];
  const float om2 = om * om;
  const float g2c = 2.f * gam[h];
  const float vv  = vco[h];

  if (wv == 0) __builtin_amdgcn_s_wait_tensorcnt(0);   // W slice landed in LDS
  __threadfence();
  grid_barrier(bar, NWG_REC);          // all y0 columns visible everywhere

  int cur = 0;
  for (int t = 0; t < T_STEPS; ++t) {
    const float* yin  = ybuf + (size_t)cur * (BATCH * N_HIDDEN);
    float*       yout = ybuf + (size_t)(cur ^ 1) * (BATCH * N_HIDDEN);

    // Prefetch next step's ext element for this lane (speculative; clamped)
    {
      const int tn = (t + 1 < T_STEPS) ? (t + 1) : t;
      const int brow0 = m0 + 8 * half;
      __builtin_prefetch(out + ((size_t)tn * BATCH + brow0) * N_HIDDEN + h, 0, 1);
    }

    // rec tile = y_full(16 rows) @ W_slice(512 x 16 cols), K stepped by 4
    v8f acc = {0.f, 0.f, 0.f, 0.f, 0.f, 0.f, 0.f, 0.f};
    const float* Arow  = yin + (size_t)(m0 + lm) * N_HIDDEN + 2 * half;
    const float* Bbase = sW + (size_t)(2 * half) * 32 + nloc;
#pragma unroll 4
    for (int k0 = 0; k0 < N_HIDDEN; k0 += 4) {
      v2f a, b;
      a.x = Arow[k0 + 0];
      a.y = Arow[k0 + 1];
      b.x = Bbase[(size_t)(k0 + 0) * 32];
      b.y = Bbase[(size_t)(k0 + 1) * 32];
      acc = __builtin_amdgcn_wmma_f32_16x16x4_f32(false, a, false, b,
                                                  (short)0, acc, false, false);
    }

    // Elementwise oscillator update for the 8 rows this lane owns
#pragma unroll
    for (int r = 0; r < 8; ++r) {
      const int brow = m0 + r + 8 * half;                       // batch index
      const size_t oidx = ((size_t)t * BATCH + brow) * N_HIDDEN + h;
      const float e  = out[oidx];                               // ext[t,b,h]
      const float x  = sX[brow * 32 + nloc];
      const float y  = sY[brow * 32 + nloc];
      const float ip = e + GAIN_REC * (acc[r] + bh + vv * x);
      const float yt = y + DT_F * (al * tanhf(ip) - om2 * x - g2c * y);
      const float xt = x + DT_F * yt;
      sX[brow * 32 + nloc] = xt;
      sY[brow * 32 + nloc] = yt;
      out[oidx] = xt;                                           // output x_t
      yout[(size_t)brow * N_HIDDEN + h] = yt;                   // publish y_t
    }

    __threadfence();
    grid_barrier(bar, NWG_REC);
    cur ^= 1;
  }
}

// -------------------------------------------------------------------------
extern "C" void kernel_launch(void* const* d_in, const int* in_sizes, int n_in,
                              void* d_out, int out_size, void* d_ws, size_t ws_size,
                              hipStream_t stream) {
  const float* inputs = (const float*)d_in[0];   // [T,B,256]
  const float* W_ih   = (const float*)d_in[1];   // [256,512]
  const float* b_ih   = (const float*)d_in[2];   // [512]
  const float* W_hh   = (const float*)d_in[3];   // [512,512]
  const float* b_hh   = (const float*)d_in[4];   // [512]
  const float* alp    = (const float*)d_in[5];
  const float* omg    = (const float*)d_in[6];
  const float* gam    = (const float*)d_in[7];
  const float* vco    = (const float*)d_in[8];
  float* out = (float*)d_out;

  // d_ws layout: [0,256) barrier ints; [256, 256 + 2*B*H*4) y ping-pong
  int*   bar  = (int*)d_ws;
  float* ybuf = (float*)((char*)d_ws + 256);

  horn_init<<<1, 1, 0, stream>>>(bar);
  horn_ext_gemm<<<16384, 256, 0, stream>>>(inputs, W_ih, b_ih, out);

  const size_t lds_bytes = (512 * 32 + 2 * 64 * 32) * sizeof(float); // 80 KB
  horn_recur<<<NWG_REC, 256, lds_bytes, stream>>>(out, W_hh, b_hh, alp, omg,
                                                  gam, vco, ybuf, bar);
}